// Conv4d_strang_33741263078093
// MI455X (gfx1250) — compile-verified
//
#include <hip/hip_runtime.h>

typedef __attribute__((ext_vector_type(2))) float v2f;
typedef __attribute__((ext_vector_type(8))) float v8f;

#define CSZ (32 * 32 * 64 * 64)   // elements per (b, ci) input channel slice

// Implicit-GEMM 4D conv on V_WMMA_F32_16X16X4_F32.
//   M = 16 output q positions, N = co (cols 0..3 valid),
//   K = 4 = the (kh,kw) 2x2 tap quad  -> A operand is ONE b64 load per lane.
//   Accumulate over ci (4) x uv-taps (9) = 36 WMMAs per tile, 2 tiles/wave.
__global__ __launch_bounds__(256)
void conv4d_strang_wmma(const float* __restrict__ x,
                        const float* __restrict__ w,
                        const float* __restrict__ bias,
                        float* __restrict__ out) {
    // B operands: Bs[(ci*9+g)*32 + lane] = float2 for this lane's K-pair.
    // lane L: N = L&15, K-pair = (L>=16 ? khw{2,3} : khw{0,1})
    __shared__ float2 Bs[36 * 32];

    const int tid = threadIdx.x;
    for (int i = tid; i < 36 * 32; i += 256) {
        const int ln  = i & 31;
        const int t   = i >> 5;       // t = ci*9 + g
        const int ci  = t / 9;
        const int g   = t - ci * 9;   // g = ku*3 + kv
        const int n   = ln & 15;
        const int kh0 = (ln >> 4) << 1;   // khw base: 0 or 2
        float b0 = 0.0f, b1 = 0.0f;
        if (n < 4) {
            // w flat: [co][ci][tap], tap = g*4 + khw
            b0 = w[(n * 4 + ci) * 36 + g * 4 + kh0];
            b1 = w[(n * 4 + ci) * 36 + g * 4 + kh0 + 1];
        }
        Bs[i] = make_float2(b0, b1);
    }
    __syncthreads();

    const int lane = tid & 31;
    const int wave = tid >> 5;
    const int job  = blockIdx.x * 8 + wave;   // 131072 jobs (b,u,v,p)

    const int p = job & 31;
    const int v = (job >> 5) & 31;
    const int u = (job >> 10) & 31;
    const int b = job >> 15;

    const int m  = lane & 15;      // A row / output q within tile
    const int kh = lane >> 4;      // which h-row this lane's K-pair uses
    const int hb = (p << 1) + kh;  // input h row for this lane
    const int wb = m << 1;         // tile0 input w base (stride-2, coalesced b64)

    // per-ci lane base pointers (row + w already folded in)
    const float* xc0 = x + (size_t)(b * 4) * CSZ + (size_t)hb * 64 + wb;
    const float* xc1 = xc0 + (size_t)CSZ;
    const float* xc2 = xc0 + (size_t)2 * CSZ;
    const float* xc3 = xc0 + (size_t)3 * CSZ;

    const v2f* bsl = (const v2f*)&Bs[lane];   // index by t*32, imm ds offsets

    const float bini = (m < 4) ? bias[m] : 0.0f;
    v8f c0 = {bini, bini, bini, bini, bini, bini, bini, bini};
    v8f c1 = c0;

    #pragma unroll
    for (int ku = 0; ku < 3; ++ku) {
        const int uu = u - 1 + ku;
        #pragma unroll
        for (int kv = 0; kv < 3; ++kv) {
            const int vv    = v - 1 + kv;
            const bool ok   = (uu >= 0) & (uu < 32) & (vv >= 0) & (vv < 32);
            const float msk = ok ? 1.0f : 0.0f;
            const int uuc   = uu < 0 ? 0 : (uu > 31 ? 31 : uu);
            const int vvc   = vv < 0 ? 0 : (vv > 31 ? 31 : vv);
            const size_t go = (size_t)(uuc * 32 + vvc) * 4096;  // h/w plane
            const int g     = ku * 3 + kv;

            const float* p0 = xc0 + go;
            const float* p1 = xc1 + go;
            const float* p2 = xc2 + go;
            const float* p3 = xc3 + go;

            // A operands: tile0 at +0, tile1 at +32 elements (imm 128B)
            v2f a00 = *(const v2f*)(p0);
            v2f a01 = *(const v2f*)(p1);
            v2f a02 = *(const v2f*)(p2);
            v2f a03 = *(const v2f*)(p3);
            v2f a10 = *(const v2f*)(p0 + 32);
            v2f a11 = *(const v2f*)(p1 + 32);
            v2f a12 = *(const v2f*)(p2 + 32);
            v2f a13 = *(const v2f*)(p3 + 32);

            // B operands for this uv group, masked for padding
            const v2f mv = {msk, msk};
            v2f b0v = bsl[(0 * 9 + g) * 32] * mv;
            v2f b1v = bsl[(1 * 9 + g) * 32] * mv;
            v2f b2v = bsl[(2 * 9 + g) * 32] * mv;
            v2f b3v = bsl[(3 * 9 + g) * 32] * mv;

            c0 = __builtin_amdgcn_wmma_f32_16x16x4_f32(false, a00, false, b0v,
                                                       (short)0, c0, false, false);
            c1 = __builtin_amdgcn_wmma_f32_16x16x4_f32(false, a10, false, b0v,
                                                       (short)0, c1, false, false);
            c0 = __builtin_amdgcn_wmma_f32_16x16x4_f32(false, a01, false, b1v,
                                                       (short)0, c0, false, false);
            c1 = __builtin_amdgcn_wmma_f32_16x16x4_f32(false, a11, false, b1v,
                                                       (short)0, c1, false, false);
            c0 = __builtin_amdgcn_wmma_f32_16x16x4_f32(false, a02, false, b2v,
                                                       (short)0, c0, false, false);
            c1 = __builtin_amdgcn_wmma_f32_16x16x4_f32(false, a12, false, b2v,
                                                       (short)0, c1, false, false);
            c0 = __builtin_amdgcn_wmma_f32_16x16x4_f32(false, a03, false, b3v,
                                                       (short)0, c0, false, false);
            c1 = __builtin_amdgcn_wmma_f32_16x16x4_f32(false, a13, false, b3v,
                                                       (short)0, c1, false, false);
        }
    }

    // D layout: VGPR r, lane L -> M = r + (L>=16 ? 8 : 0), N = L&15.
    if (m < 4) {
        const int mrow = kh << 3;   // 0 or 8
        float* yp = out
            + (((((size_t)b * 4 + m) * 32 + u) * 32 + v) * 32 + p) * 32
            + mrow;
        #pragma unroll
        for (int r = 0; r < 8; ++r) yp[r] = c0[r];        // tile0: q = mrow+r
        #pragma unroll
        for (int r = 0; r < 8; ++r) yp[16 + r] = c1[r];   // tile1: q = 16+mrow+r
    }
}

extern "C" void kernel_launch(void* const* d_in, const int* in_sizes, int n_in,
                              void* d_out, int out_size, void* d_ws, size_t ws_size,
                              hipStream_t stream) {
    const float* x    = (const float*)d_in[0];
    const float* w    = (const float*)d_in[1];
    const float* bias = (const float*)d_in[2];
    float* out        = (float*)d_out;

    // 4*32*32*32 = 131072 wave-jobs, 8 waves (256 threads) per block
    dim3 grid(16384), block(256);
    conv4d_strang_wmma<<<grid, block, 0, stream>>>(x, w, bias, out);
}